// GraphAttention_52484500357570
// MI455X (gfx1250) — compile-verified
//
#include <hip/hip_runtime.h>
#include <hip/hip_bf16.h>

typedef float v2f __attribute__((ext_vector_type(2)));
typedef float v8f __attribute__((ext_vector_type(8)));

#define B_   2
#define N_   2048
#define FIN  256
#define FOUT 64
#define H_   4
#define SLOPE 0.2f
#define NEG_BIG (-1e10f)

#if defined(__has_builtin)
# if __has_builtin(__builtin_amdgcn_global_load_async_to_lds_b128)
#  define HAVE_ASYNC_LDS 1
# endif
# if __has_builtin(__builtin_amdgcn_s_wait_asynccnt)
#  define HAVE_WAIT_ASYNC 1
# endif
#endif

#ifdef HAVE_ASYNC_LDS
typedef int a_v4i __attribute__((vector_size(16)));
typedef __attribute__((address_space(1))) a_v4i* g128p;
typedef __attribute__((address_space(3))) a_v4i* l128p;
typedef __attribute__((address_space(1))) int*   g32p;
typedef __attribute__((address_space(3))) int*   l32p;

__device__ __forceinline__ void async_copy_b128(const void* g, void* l) {
  __builtin_amdgcn_global_load_async_to_lds_b128((g128p)g, (l128p)l, 0, 0);
}
__device__ __forceinline__ void async_copy_b32(const void* g, void* l) {
  __builtin_amdgcn_global_load_async_to_lds_b32((g32p)g, (l32p)l, 0, 0);
}
#endif

// ---------------------------------------------------------------------------
// Kernel 1: h_prime[bh, n, o] = sum_k h[b,n,k] * w[head,k,o]
// One wave (32 threads) per 16-row x 64-col tile. fp32 WMMA 16x16x4.
// ---------------------------------------------------------------------------
__global__ __launch_bounds__(32) void k_hprime(const float* __restrict__ h,
                                               const float* __restrict__ w,
                                               float* __restrict__ hp) {
  const int ntiles = N_ / 16;
  int wid   = blockIdx.x;
  int ntile = wid % ntiles;
  int bh    = wid / ntiles;          // b*H + head
  int head  = bh % H_;
  int b     = bh / H_;

  int lane = threadIdx.x;
  int m = lane & 15;
  int g = lane >> 4;

  int row = ntile * 16 + m;
  const float* hrow = h + ((size_t)b * N_ + row) * FIN;
  const float* wh   = w + (size_t)head * FIN * FOUT;

  v8f acc[4] = {};
  for (int k0 = 0; k0 < FIN; k0 += 4) {
    int ka = k0 + 2 * g;
    float2 av = *(const float2*)(hrow + ka);
    v2f a; a[0] = av.x; a[1] = av.y;
    const float* wk = wh + (size_t)ka * FOUT;
#pragma unroll
    for (int t = 0; t < 4; ++t) {
      int n = t * 16 + m;
      v2f bv; bv[0] = wk[n]; bv[1] = wk[FOUT + n];
      acc[t] = __builtin_amdgcn_wmma_f32_16x16x4_f32(
          false, a, false, bv, (short)0, acc[t], false, false);
    }
  }

  // lane holds D[r + 8g][t*16 + m] in acc[t][r]
  float* hpb = hp + ((size_t)bh * N_ + (size_t)ntile * 16) * FOUT;
#pragma unroll
  for (int t = 0; t < 4; ++t)
#pragma unroll
    for (int r = 0; r < 8; ++r)
      hpb[(size_t)(r + 8 * g) * FOUT + t * 16 + m] = acc[t][r];
}

// ---------------------------------------------------------------------------
// Kernel 2: attn_src/dst[bh,n] = sum_o tanh(hp[bh,n,o]) * a_{src,dst}[head,o]
// One wave per row, shfl-xor reduction.
// ---------------------------------------------------------------------------
__global__ __launch_bounds__(256) void k_attnvec(const float* __restrict__ hp,
                                                 const float* __restrict__ a_src,
                                                 const float* __restrict__ a_dst,
                                                 float* __restrict__ attn_src,
                                                 float* __restrict__ attn_dst) {
  int wid  = threadIdx.x >> 5;
  int lane = threadIdx.x & 31;
  size_t row = (size_t)blockIdx.x * 8 + wid;   // 0 .. B*H*N-1
  int bh   = (int)(row / N_);
  int head = bh % H_;

  const float* hr = hp + row * FOUT;
  float t0 = tanhf(hr[lane]);
  float t1 = tanhf(hr[lane + 32]);
  const float* as = a_src + head * FOUT;
  const float* ad = a_dst + head * FOUT;
  float s = t0 * as[lane] + t1 * as[lane + 32];
  float d = t0 * ad[lane] + t1 * ad[lane + 32];
#pragma unroll
  for (int off = 16; off; off >>= 1) {
    s += __shfl_xor(s, off, 32);
    d += __shfl_xor(d, off, 32);
  }
  if (lane == 0) { attn_src[row] = s; attn_dst[row] = d; }
}

// ---------------------------------------------------------------------------
// Kernel 3: flash-style masked attention + P @ h_prime, fp32 WMMA.
// Block = 256 threads = 8 waves, each wave owns 16 rows; j-tiles of 64.
// V tile staged global->LDS via async-LDS loads (ASYNCcnt) when available.
// ---------------------------------------------------------------------------
__global__ __launch_bounds__(256) void k_gat(const float* __restrict__ hp,
                                             const int*   __restrict__ adj,
                                             const float* __restrict__ asrc,
                                             const float* __restrict__ adst,
                                             const float* __restrict__ bias,
                                             float* __restrict__ out) {
  __shared__ float Vt[64][68];   // pad 68 -> conflict-free half-wave reads
  __shared__ float Dv[64];       // attn_dst slice for this j-tile

  const int blocksPerBH = N_ / 128;             // 16
  int bh    = blockIdx.x / blocksPerBH;
  int rtile = blockIdx.x % blocksPerBH;
  int head  = bh % H_;
  int b     = bh / H_;

  int wid  = threadIdx.x >> 5;
  int lane = threadIdx.x & 31;
  int m = lane & 15;
  int g = lane >> 4;

  int rowbase = rtile * 128 + wid * 16;
  int i = rowbase + m;                          // row this lane scores

  const float* hpb = hp + (size_t)bh * N_ * FOUT;
  float src_i = asrc[(size_t)bh * N_ + i];
  const int* adjrow = adj + ((size_t)b * N_ + i) * N_ * H_ + head;
  const float* dstv = adst + (size_t)bh * N_;

  float m_run = -INFINITY;
  float l_run = 0.f;
  v8f acc[4] = {};

  for (int j0 = 0; j0 < N_; j0 += 64) {
    __syncthreads();
    // ---- stage V tile = hp[bh, j0:j0+64, 0:64] and dst slice into LDS ----
#ifdef HAVE_ASYNC_LDS
    for (int x = threadIdx.x; x < 64 * 16; x += 256) {
      int jj = x >> 4;
      int oo = (x & 15) * 4;
      async_copy_b128(&hpb[(size_t)(j0 + jj) * FOUT + oo], &Vt[jj][oo]);
    }
    if (threadIdx.x < 64)
      async_copy_b32(&dstv[j0 + threadIdx.x], &Dv[threadIdx.x]);
# ifdef HAVE_WAIT_ASYNC
    __builtin_amdgcn_s_wait_asynccnt(0);
# else
    asm volatile("s_wait_asynccnt 0" ::: "memory");
# endif
#else
    for (int x = threadIdx.x; x < 64 * 16; x += 256) {
      int jj = x >> 4;
      int oo = (x & 15) * 4;
      *(float4*)&Vt[jj][oo] = *(const float4*)&hpb[(size_t)(j0 + jj) * FOUT + oo];
    }
    if (threadIdx.x < 64) Dv[threadIdx.x] = dstv[j0 + threadIdx.x];
#endif
    __syncthreads();

    // prefetch next adj tile (the only HBM-bound stream)
    if (j0 + 64 < N_)
      __builtin_prefetch((const void*)(adjrow + (size_t)(j0 + 64 + 2 * g) * H_), 0, 1);

    // scores for this lane's j subset (j mod 4 in {2g, 2g+1}), A-operand layout
    float p[32];
    float mloc = -INFINITY;
#pragma unroll
    for (int kk = 0; kk < 16; ++kk) {
#pragma unroll
      for (int e = 0; e < 2; ++e) {
        int j = 4 * kk + 2 * g + e;
        float sc = src_i + Dv[j];
        sc = (sc >= 0.f) ? sc : SLOPE * sc;
        int av = adjrow[(size_t)(j0 + j) * H_];
        sc = (av > 0) ? sc : NEG_BIG;
        p[2 * kk + e] = sc;
        mloc = fmaxf(mloc, sc);
      }
    }
    mloc = fmaxf(mloc, __shfl_xor(mloc, 16, 32));
    float m_new = fmaxf(m_run, mloc);
    float scale = __expf(m_run - m_new);        // exp(-inf)=0 on first tile
    float tsum = 0.f;
#pragma unroll
    for (int q = 0; q < 32; ++q) {
      p[q] = __expf(p[q] - m_new);
      tsum += p[q];
    }
    tsum += __shfl_xor(tsum, 16, 32);
    l_run = l_run * scale + tsum;
    m_run = m_new;

    // rescale accumulators: row of acc[t][r] is r + 8g; stats live in lane (row&15)
#pragma unroll
    for (int r = 0; r < 8; ++r) {
      float sr = __shfl(scale, r + 8 * g, 32);
#pragma unroll
      for (int t = 0; t < 4; ++t) acc[t][r] *= sr;
    }

    // acc += P(16x64) @ V(64x64) via fp32 WMMA, K-steps of 4
#pragma unroll
    for (int kk = 0; kk < 16; ++kk) {
      v2f a2; a2[0] = p[2 * kk]; a2[1] = p[2 * kk + 1];
      int k = 4 * kk + 2 * g;
#pragma unroll
      for (int t = 0; t < 4; ++t) {
        int n = t * 16 + m;
        v2f bv; bv[0] = Vt[k][n]; bv[1] = Vt[k + 1][n];
        acc[t] = __builtin_amdgcn_wmma_f32_16x16x4_f32(
            false, a2, false, bv, (short)0, acc[t], false, false);
      }
    }
  }

  // epilogue: divide by row sum, add bias, write [B, N, H*FOUT]
#pragma unroll
  for (int r = 0; r < 8; ++r) {
    float lr  = __shfl(l_run, r + 8 * g, 32);
    float inv = 1.0f / lr;
    int gi = rowbase + r + 8 * g;
#pragma unroll
    for (int t = 0; t < 4; ++t) {
      int o = t * 16 + m;
      out[(((size_t)b * N_ + gi) * H_ + head) * FOUT + o] = acc[t][r] * inv + bias[o];
    }
  }
}

// ---------------------------------------------------------------------------
extern "C" void kernel_launch(void* const* d_in, const int* in_sizes, int n_in,
                              void* d_out, int out_size, void* d_ws, size_t ws_size,
                              hipStream_t stream) {
  const float* h     = (const float*)d_in[0];
  const int*   adj   = (const int*)  d_in[1];
  const float* w     = (const float*)d_in[2];
  const float* a_src = (const float*)d_in[3];
  const float* a_dst = (const float*)d_in[4];
  const float* bias  = (const float*)d_in[5];
  float* out = (float*)d_out;

  const size_t HP_ELEMS  = (size_t)B_ * H_ * N_ * FOUT;  // 1M floats
  const size_t VEC_ELEMS = (size_t)B_ * H_ * N_;         // 16K floats

  float* hp       = (float*)d_ws;
  float* attn_src = hp + HP_ELEMS;
  float* attn_dst = attn_src + VEC_ELEMS;

  // Kernel 1: h_prime GEMM (one wave per 16x64 tile)
  k_hprime<<<B_ * H_ * (N_ / 16), 32, 0, stream>>>(h, w, hp);

  // Kernel 2: attention source/dest vectors (one wave per row)
  k_attnvec<<<(B_ * H_ * N_) / 8, 256, 0, stream>>>(hp, a_src, a_dst,
                                                    attn_src, attn_dst);

  // Kernel 3: masked softmax attention + output GEMM
  k_gat<<<B_ * H_ * (N_ / 128), 256, 0, stream>>>(hp, adj, attn_src, attn_dst,
                                                  bias, out);
}